// ADJlayer_63419487093320
// MI455X (gfx1250) — compile-verified
//
#include <hip/hip_runtime.h>
#include <stdint.h>

// out[b, r, c] = 1 - sigmoid(|r - c| + BETA), broadcast over b.
// Pure store-bandwidth problem: 512 MiB of output, ~23 us floor at 23.3 TB/s.
// Strategy: one workgroup per row; build the row once in LDS (values depend
// only on |r-c|), then stream it to all 8 batch copies with CDNA5 async
// store-from-LDS (global_store_async_from_lds_b128, ASYNCcnt-tracked) --
// no VGPR staging, 8x reuse of each computed value.

#define N    4096
#define BETA 0.8f
#define BLK  256

// Async builtins take v4i pointers in target address spaces 1 (global)
// and 3 (LDS).
typedef int v4i __attribute__((vector_size(16)));
typedef __attribute__((address_space(1))) v4i gv4i;
typedef __attribute__((address_space(3))) v4i lv4i;

__device__ __forceinline__ gv4i* to_global_v4(void* p) {
  return (gv4i*)(unsigned long long)(uintptr_t)p;
}
__device__ __forceinline__ lv4i* to_lds_v4(void* p) {
  // generic shared pointer: low 32 bits are the LDS byte offset
  return (lv4i*)(unsigned)(uintptr_t)p;
}

__global__ void __launch_bounds__(BLK)
adj_rows_kernel(float* __restrict__ out, int bs) {
  __shared__ __align__(16) float row[N];
  const int r   = blockIdx.x;
  const int tid = threadIdx.x;

  // ---- Build one output row in LDS: row[c] = 1 - sigmoid(|r-c| + BETA) ----
  // 1 - sigmoid(x) = t/(1+t) with t = exp(-x): v_exp + v_rcp + v_mul,
  // avoids the IEEE divide expansion. t -> 0 at large d gives exactly 0,
  // matching the reference's saturation.
  for (int c0 = tid * 4; c0 < N; c0 += BLK * 4) {
    float4 v;
#pragma unroll
    for (int i = 0; i < 4; ++i) {
      const int   c = c0 + i;
      const int   d = (r >= c) ? (r - c) : (c - r);
      const float x = (float)d + BETA;
      const float t = __expf(-x);
      ((float*)&v)[i] = t * __builtin_amdgcn_rcpf(1.0f + t);
    }
    *reinterpret_cast<float4*>(&row[c0]) = v;
  }
  __syncthreads();

  // ---- Stream the row to every batch copy, directly from LDS ----
  for (int b = 0; b < bs; ++b) {
    float* dst = out + ((size_t)b * N + (size_t)r) * (size_t)N;
#pragma unroll
    for (int k = 0; k < N / (BLK * 4); ++k) {
      const int e = k * BLK * 4 + tid * 4;
#if __has_builtin(__builtin_amdgcn_global_store_async_from_lds_b128)
      __builtin_amdgcn_global_store_async_from_lds_b128(
          to_global_v4(dst + e), to_lds_v4(&row[e]), /*offset=*/0, /*cpol=*/0);
#else
      asm volatile("global_store_async_from_lds_b128 %0, %1, off"
                   :
                   : "v"((unsigned long long)(uintptr_t)(dst + e)),
                     "v"((unsigned)(uintptr_t)(&row[e]))
                   : "memory");
#endif
    }
  }

  // Drain ASYNCcnt before wave termination.
#if __has_builtin(__builtin_amdgcn_s_wait_asynccnt)
  __builtin_amdgcn_s_wait_asynccnt(0);
#else
  asm volatile("s_wait_asynccnt 0" ::: "memory");
#endif
}

extern "C" void kernel_launch(void* const* d_in, const int* in_sizes, int n_in,
                              void* d_out, int out_size, void* d_ws, size_t ws_size,
                              hipStream_t stream) {
  (void)d_in; (void)in_sizes; (void)n_in; (void)d_ws; (void)ws_size;
  float* out = (float*)d_out;
  const int bs = out_size / (N * N);  // = 8 for the reference shapes
  adj_rows_kernel<<<N, BLK, 0, stream>>>(out, bs);
}